// RadiusGraph_128849019529
// MI455X (gfx1250) — compile-verified
//
#include <hip/hip_runtime.h>

typedef __attribute__((ext_vector_type(2))) float v2f;
typedef __attribute__((ext_vector_type(8))) float v8f;

#define RADIUS2   2.56f     // 1.6^2
#define KNN       32
#define CAP       64        // candidate cap per query (E[neighbors] ~ 8.8)
#define NWAVES    4
#define BSCALE    16.0f     // batch-id -> 4th coord; mismatch penalty >= 256 >> r^2

// Block = 4 waves sharing ONE 16-query tile; waves split the ref sweep.
// Each wave computes 16x16 distance tiles with V_WMMA_F32_16X16X4_F32:
//   A = -2 * (16 queries x 4 coords), B = (4 coords x 16 refs),
//   D = -2 q.r ; d2 = |q4|^2 + |r4|^2 + D.
// Candidates appended to block-shared LDS lists; phase 2 sorts (d2, idx).
__global__ __launch_bounds__(NWAVES * 32)
void radius_graph_wmma(const float* __restrict__ ref,    // [N,4] (b,x,y,z)
                       const float* __restrict__ query,  // [M,4] (b,x,y,z)
                       int* __restrict__ out,            // [2, M*KNN]
                       int N, int M)                     // N % (16*NWAVES)==0, M % 16==0
{
    __shared__ int   s_cnt[16];
    __shared__ float s_d2 [16][CAP];
    __shared__ int   s_id [16][CAP];

    const int tid  = threadIdx.x;
    const int wave = tid >> 5;
    const int lane = tid & 31;
    const int l16  = lane & 15;
    const bool hiH = (lane >= 16);

    if (tid < 16) s_cnt[tid] = 0;
    __syncthreads();

    const int m0 = blockIdx.x * 16;

    // ---- Query tile (same for all waves in block).
    const float4 qp = ((const float4*)query)[m0 + l16];
    const float qx = qp.y, qy = qp.z, qz = qp.w, qw = qp.x * BSCALE;
    const float qq = qx*qx + qy*qy + qz*qz + qw*qw;

    // A fragment (16x4 f32): lanes 0-15 hold K=0,1 ; lanes 16-31 hold K=2,3.
    v2f a;
    a.x = hiH ? (-2.0f * qz) : (-2.0f * qx);
    a.y = hiH ? (-2.0f * qw) : (-2.0f * qy);

    // Broadcast |q|^2 per C-row: C VGPR v holds M=v (lanes 0-15) / v+8 (lanes 16-31).
    float qqA[8];
#pragma unroll
    for (int v = 0; v < 8; ++v) {
        float lo = __shfl(qq, v, 32);
        float hi = __shfl(qq, v + 8, 32);
        qqA[v] = hiH ? hi : lo;
    }

    // ---- Ref sweep: waves stride the tiles; loads software-pipelined.
    const int step = 16 * NWAVES;
    int n0 = wave * 16;
    float4 rp = ((const float4*)ref)[n0 + l16];

#pragma unroll 2
    for (; n0 < N; n0 += step) {
        int n1 = n0 + step; if (n1 >= N) n1 = n0;          // dummy reload on last iter
        const float4 rp_next = ((const float4*)ref)[n1 + l16];

        const float rx = rp.y, ry = rp.z, rz = rp.w, rw = rp.x * BSCALE;
        const float rr = rx*rx + ry*ry + rz*rz + rw*rw;

        // B fragment (4x16 f32): lanes 0-15 hold K=0,1 ; lanes 16-31 hold K=2,3.
        v2f b;
        b.x = hiH ? rz : rx;
        b.y = hiH ? rw : ry;

        v8f c = {};
        c = __builtin_amdgcn_wmma_f32_16x16x4_f32(
                false, a, false, b, (short)0, c, false, false);

        float d2v[8];
        float dmin = 1e30f;
#pragma unroll
        for (int v = 0; v < 8; ++v) {
            d2v[v] = qqA[v] + rr + c[v];       // |q|^2+|r|^2-2q.r (+batch penalty)
            dmin   = fminf(dmin, d2v[v]);
        }

        // ~87% of tiles have zero candidates: one wave-wide test, one branch.
        if (__any(dmin <= RADIUS2)) {
#pragma unroll
            for (int v = 0; v < 8; ++v) {
                if (d2v[v] <= RADIUS2) {
                    const int q    = v + (hiH ? 8 : 0);
                    const int slot = atomicAdd(&s_cnt[q], 1);
                    if (slot < CAP) {
                        s_d2[q][slot] = d2v[v];
                        s_id[q][slot] = n0 + l16;   // C lane's N == its loaded ref
                    }
                }
            }
        }
        rp = rp_next;
    }
    __syncthreads();

    // ---- Phase 2: per-query top-KNN, deterministic tie-break on (d2, idx).
    if (tid < 16) {
        const int q  = tid;
        const int qg = m0 + q;
        int n = s_cnt[q]; if (n > CAP) n = CAP;
        const int ksel = (n < KNN) ? n : KNN;
        float* dz = s_d2[q];
        int*   iz = s_id[q];

        for (int k = 0; k < ksel; ++k) {
            int best = k; float bd = dz[k]; int bi = iz[k];
            for (int j = k + 1; j < n; ++j) {
                const float dj = dz[j];
                const int   ij = iz[j];
                if (dj < bd || (dj == bd && ij < bi)) { bd = dj; bi = ij; best = j; }
            }
            if (best != k) {
                const float td = dz[k]; dz[k] = dz[best]; dz[best] = td;
                const int   ti = iz[k]; iz[k] = iz[best]; iz[best] = ti;
            }
        }

        const long long base = (long long)qg * KNN;
        const long long MK   = (long long)M * KNN;
        for (int k = 0; k < KNN; ++k) {
            out[base + k]      = (k < ksel) ? iz[k] : -1;   // row 0: ref indices
            out[MK + base + k] = (k < ksel) ? qg    : -1;   // row 1: query indices
        }
    }
}

extern "C" void kernel_launch(void* const* d_in, const int* in_sizes, int n_in,
                              void* d_out, int out_size, void* d_ws, size_t ws_size,
                              hipStream_t stream) {
    const float* ref   = (const float*)d_in[0];   // ref_bxyz   [N,4]
    const float* query = (const float*)d_in[1];   // query_bxyz [M,4]
    int* out = (int*)d_out;                        // edges [2, M*KNN], int32

    const int N = in_sizes[0] / 4;
    const int M = in_sizes[1] / 4;

    const int blocks = (M + 15) / 16;              // M=16384 -> 1024 blocks
    radius_graph_wmma<<<blocks, NWAVES * 32, 0, stream>>>(ref, query, out, N, M);
}